// NodeBlockCUGO_31705448579495
// MI455X (gfx1250) — compile-verified
//
#include <hip/hip_runtime.h>

// ---------------------------------------------------------------------------
// NodeBlock (GNN message-passing MLP) for MI455X / gfx1250, wave32.
//
// Roofline: ~5 GB of unavoidable HBM traffic (efeat read + passthrough copy)
// vs ~40 GFLOP of GEMM -> strongly memory bound, so GEMMs run in full f32 via
// V_WMMA_F32_16X16X4_F32 (precision-exact path, matrix pipe still engaged).
// ---------------------------------------------------------------------------

#define N_NODES 100000
#define N_EDGES 1600000
#define D       256
#define H       256
#define DIN     512          // concat(agg, nfeat)
#define LN_EPS  1e-5f

#define EDGES_PER_BLOCK 32
#define XS_STRIDE 516        // 516 % 64 == 4 -> conflict-free ds_load_b64
#define HS_STRIDE 260        // 260 % 64 == 4

typedef __attribute__((ext_vector_type(2))) float v2f;
typedef __attribute__((ext_vector_type(8))) float v8f;

// ---------------------------------------------------------------------------
// Kernel 1: zero the aggregation buffer (aliased with nfeat_new output region)
// ---------------------------------------------------------------------------
__global__ void zero_agg_kernel(float4* __restrict__ p) {
  int i = blockIdx.x * blockDim.x + threadIdx.x;   // exact grid, no tail
  p[i] = make_float4(0.f, 0.f, 0.f, 0.f);
}

// ---------------------------------------------------------------------------
// Kernel 2: segment-sum (scatter-add) of edge features into dst nodes.
// edge_dst is sorted, so run-compress equal-dst runs in registers and only
// touch memory atomically at run boundaries (~N_EDGES/runlen atomics).
// Thread t owns feature column t (coalesced 1 KB rows).
// ---------------------------------------------------------------------------
__global__ __launch_bounds__(256)
void scatter_add_kernel(const float* __restrict__ efeat,
                        const int*   __restrict__ edge_dst,
                        float*       __restrict__ agg) {
  const int t = threadIdx.x;                       // column 0..255
  const long base = (long)blockIdx.x * EDGES_PER_BLOCK;

  int   prev = edge_dst[base];
  float acc  = 0.f;
  for (int i = 0; i < EDGES_PER_BLOCK; ++i) {
    const long e = base + i;
    const int  d = edge_dst[e];                    // uniform -> scalar load
    if (d != prev) {
      atomicAdd(&agg[(long)prev * D + t], acc);    // global_atomic_add_f32
      acc  = 0.f;
      prev = d;
    }
    acc += efeat[e * (long)D + t];
  }
  atomicAdd(&agg[(long)prev * D + t], acc);
}

// ---------------------------------------------------------------------------
// Kernel 3: fused MLP + LayerNorm + residual for a 16-node tile.
// 256 threads = 8 wave32s; each wave computes two 16x16 output tiles per GEMM
// with V_WMMA_F32_16X16X4_F32.  A-fragments come from padded LDS, B-fragments
// stream from W1/W2 (L2-resident).  agg aliases the output region: each block
// stages its 16 rows into LDS before overwriting them.
// ---------------------------------------------------------------------------
__global__ __launch_bounds__(256)
void mlp_fused_kernel(const float* __restrict__ agg,
                      const float* __restrict__ nfeat,
                      const float* __restrict__ w1, const float* __restrict__ b1,
                      const float* __restrict__ w2, const float* __restrict__ b2,
                      const float* __restrict__ ln_g, const float* __restrict__ ln_b,
                      float* __restrict__ out_nodes) {
  __shared__ float Xs[16 * XS_STRIDE];   // [16 x 512] concat(agg, nfeat), padded
  __shared__ float Hs[16 * HS_STRIDE];   // [16 x 256] SiLU(GEMM1)
  __shared__ float Os[16 * HS_STRIDE];   // [16 x 256] GEMM2 + b2
  __shared__ float mu_s[16];
  __shared__ float rs_s[16];

  const int tid   = threadIdx.x;
  const int wave  = tid >> 5;
  const int lane  = tid & 31;
  const int node0 = blockIdx.x * 16;     // N_NODES = 6250 * 16, exact

  // ---- stage X = [agg | nfeat] into LDS (coalesced) ----
  for (int idx = tid; idx < 16 * DIN; idx += 256) {
    const int m = idx >> 9;              // row 0..15
    const int k = idx & (DIN - 1);       // col 0..511
    const float v = (k < D) ? agg  [(size_t)(node0 + m) * D + k]
                            : nfeat[(size_t)(node0 + m) * D + (k - D)];
    Xs[m * XS_STRIDE + k] = v;
  }
  __syncthreads();

  // WMMA f32 16x16x4 operand mapping (ISA 7.12.2):
  //   A (16x4):  lanes 0-15 -> M=lane,   VGPRs = K+0,K+1
  //              lanes 16-31-> M=lane-16,VGPRs = K+2,K+3
  //   B (4x16):  lanes 0-15 -> N=lane,   rows K+0,K+1 ; lanes 16-31 rows K+2,K+3
  //   C/D:       VGPR r -> row r (lanes 0-15) / row r+8 (lanes 16-31), col=lane%16
  const int mrow  = lane & 15;
  const int khalf = (lane >> 4) << 1;    // 0 or 2
  const int nsub  = lane & 15;
  const int rbase = (lane >> 4) << 3;    // 0 or 8

  // ---- GEMM1: [16 x 512] @ [512 x 256] -> SiLU -> Hs ----
  for (int nt = 0; nt < 2; ++nt) {
    const int ncol0 = (wave * 2 + nt) * 16;
    const int bcol  = ncol0 + nsub;
    v8f c = {0.f, 0.f, 0.f, 0.f, 0.f, 0.f, 0.f, 0.f};
    const float* __restrict__ xrow = &Xs[mrow * XS_STRIDE + khalf];
    const float* __restrict__ wcol = &w1[(size_t)khalf * H + bcol];
    for (int k = 0; k < DIN; k += 4) {
      v2f a = *(const v2f*)(xrow + k);                  // ds_load_b64, no conflicts
      v2f b;
      b.x = wcol[(size_t)k * H];
      b.y = wcol[(size_t)k * H + H];
      c = __builtin_amdgcn_wmma_f32_16x16x4_f32(false, a, false, b,
                                                (short)0, c, false, false);
    }
    const float bv = b1[bcol];
#pragma unroll
    for (int r = 0; r < 8; ++r) {
      const float x = c[r] + bv;
      Hs[(rbase + r) * HS_STRIDE + bcol] = x / (1.f + __expf(-x));  // SiLU
    }
  }
  __syncthreads();

  // ---- GEMM2: [16 x 256] @ [256 x 256] + b2 -> Os ----
  for (int nt = 0; nt < 2; ++nt) {
    const int ncol0 = (wave * 2 + nt) * 16;
    const int bcol  = ncol0 + nsub;
    v8f c = {0.f, 0.f, 0.f, 0.f, 0.f, 0.f, 0.f, 0.f};
    const float* __restrict__ hrow = &Hs[mrow * HS_STRIDE + khalf];
    const float* __restrict__ wcol = &w2[(size_t)khalf * D + bcol];
    for (int k = 0; k < H; k += 4) {
      v2f a = *(const v2f*)(hrow + k);
      v2f b;
      b.x = wcol[(size_t)k * D];
      b.y = wcol[(size_t)k * D + D];
      c = __builtin_amdgcn_wmma_f32_16x16x4_f32(false, a, false, b,
                                                (short)0, c, false, false);
    }
    const float bv = b2[bcol];
#pragma unroll
    for (int r = 0; r < 8; ++r) {
      Os[(rbase + r) * HS_STRIDE + bcol] = c[r] + bv;
    }
  }
  __syncthreads();

  // ---- LayerNorm row statistics (16 rows, one thread each) ----
  if (tid < 16) {
    const float* __restrict__ orow = &Os[tid * HS_STRIDE];
    float s = 0.f, s2 = 0.f;
    for (int j = 0; j < D; ++j) {
      const float v = orow[j];
      s  += v;
      s2 += v * v;
    }
    const float m   = s * (1.f / D);
    const float var = s2 * (1.f / D) - m * m;
    mu_s[tid] = m;
    rs_s[tid] = rsqrtf(var + LN_EPS);
  }
  __syncthreads();

  // ---- normalize * g + b, add residual (nfeat half of Xs), store ----
  const int col = tid;                               // 0..255
  const float g  = ln_g[col];
  const float bb = ln_b[col];
  for (int r = 0; r < 16; ++r) {
    const float v = (Os[r * HS_STRIDE + col] - mu_s[r]) * rs_s[r] * g + bb
                  + Xs[r * XS_STRIDE + D + col];     // residual = nfeat
    out_nodes[(size_t)(node0 + r) * D + col] = v;
  }
}

// ---------------------------------------------------------------------------
extern "C" void kernel_launch(void* const* d_in, const int* in_sizes, int n_in,
                              void* d_out, int out_size, void* d_ws, size_t ws_size,
                              hipStream_t stream) {
  const float* efeat    = (const float*)d_in[0];
  const float* nfeat    = (const float*)d_in[1];
  const int*   edge_dst = (const int*)  d_in[2];
  const float* w1       = (const float*)d_in[3];
  const float* b1       = (const float*)d_in[4];
  const float* w2       = (const float*)d_in[5];
  const float* b2       = (const float*)d_in[6];
  const float* ln_g     = (const float*)d_in[7];
  const float* ln_b     = (const float*)d_in[8];

  float* out = (float*)d_out;
  // Output layout: [ efeat passthrough | nfeat_new ].  The nfeat_new region
  // doubles as the aggregation buffer (safe: each MLP block stages its rows
  // into LDS before overwriting them in place).
  float* agg = out + (size_t)N_EDGES * D;

  // 1) efeat passthrough (independent region; overlaps other work)
  hipMemcpyAsync(out, efeat, (size_t)N_EDGES * D * sizeof(float),
                 hipMemcpyDeviceToDevice, stream);

  // 2) zero agg (must happen every call; buffer holds last call's output)
  zero_agg_kernel<<<(N_NODES * D / 4) / 256, 256, 0, stream>>>((float4*)agg);

  // 3) sorted scatter-add with run compression
  scatter_add_kernel<<<N_EDGES / EDGES_PER_BLOCK, 256, 0, stream>>>(
      efeat, edge_dst, agg);

  // 4) fused MLP + LayerNorm + residual (writes agg region in place)
  mlp_fused_kernel<<<N_NODES / 16, 256, 0, stream>>>(
      agg, nfeat, w1, b1, w2, b2, ln_g, ln_b, agg);
}